// YoloLoss_5549097746611
// MI455X (gfx1250) — compile-verified
//
#include <hip/hip_runtime.h>
#include <hip/hip_bf16.h>
#include <math.h>
#include <stdint.h>

#define N_PRED 8192
#define N_GT   256
#define BATCH  16
#define BLOCKS_PER_IMG 32
#define TPB 256

typedef float v2f __attribute__((ext_vector_type(2)));
typedef float v8f __attribute__((ext_vector_type(8)));

// ---------------------------------------------------------------------------
// Kernel 0: zero the per-image accumulators (err, cnt, bce) x 16 images
// ---------------------------------------------------------------------------
__global__ void yolo_zero_ws(float* ws, int n) {
    int i = threadIdx.x;
    if (i < n) ws[i] = 0.0f;
}

// ---------------------------------------------------------------------------
// Kernel 1: per-prediction IoU argmax + partial sums.
//   grid = BATCH * BLOCKS_PER_IMG blocks, TPB threads; 1 prediction / thread.
//   GT boxes staged to LDS via gfx1250 async global->LDS load.
// ---------------------------------------------------------------------------
__global__ __launch_bounds__(TPB) void yolo_main(
    const float* __restrict__ y_hat,   // (B, N_PRED, 4) xywh
    const float* __restrict__ y,       // (B, N_GT, 4)   xywh
    const float* __restrict__ obj,     // (B, N_PRED)
    float* __restrict__ acc)           // (B, 3): err_sum, cnt, bce_sum
{
    __shared__ float4 rawGT[N_GT];     // raw xywh (needed for sq-err vs matched gt)
    __shared__ float4 gbox[N_GT];      // x1,y1,x2,y2
    __shared__ float  gar[N_GT];       // gt area
    __shared__ float  redE[8], redC[8], redB[8];

    const int tid = threadIdx.x;
    const int img = blockIdx.x / BLOCKS_PER_IMG;
    const int blk = blockIdx.x % BLOCKS_PER_IMG;

    // ---- async stage: each thread pulls one GT float4 straight into LDS ----
    {
        const float4* gsrc = (const float4*)(y + (size_t)img * N_GT * 4) + tid;
        unsigned int  ldsOff = (unsigned int)(size_t)(void*)(&rawGT[tid]);
        unsigned long long ga = (unsigned long long)(size_t)(const void*)gsrc;
        asm volatile("global_load_async_to_lds_b128 %0, %1, off"
                     :: "v"(ldsOff), "v"(ga) : "memory");
        asm volatile("s_wait_asynccnt 0x0" ::: "memory");
        float4 g = rawGT[tid];                 // own lane's entry: safe post-wait
        float hw = 0.5f * g.z, hh = 0.5f * g.w;
        gbox[tid] = make_float4(g.x - hw, g.y - hh, g.x + hw, g.y + hh);
        gar[tid]  = g.z * g.w;
    }
    __syncthreads();

    // ---- my prediction ----
    const int p = blk * TPB + tid;
    const float4 pb = ((const float4*)(y_hat + (size_t)img * N_PRED * 4))[p];
    const float  s  = obj[(size_t)img * N_PRED + p];
    const float phw = 0.5f * pb.z, phh = 0.5f * pb.w;
    const float px1 = pb.x - phw, py1 = pb.y - phh;
    const float px2 = pb.x + phw, py2 = pb.y + phh;
    const float parea = pb.z * pb.w;

    // division-free running argmax: best iou = bi/bu (unions > 0)
    float bi = -1.0f, bu = 1.0f;
    int   bidx = 0;
#pragma unroll 4
    for (int j = 0; j < N_GT; ++j) {
        float4 g = gbox[j];                     // LDS broadcast (same addr all lanes)
        float ga = gar[j];
        float iw = fminf(px2, g.z) - fmaxf(px1, g.x);
        float ih = fminf(py2, g.w) - fmaxf(py1, g.y);
        iw = fmaxf(iw, 0.0f);
        ih = fmaxf(ih, 0.0f);
        float inter = iw * ih;
        float u = parea + ga - inter;
        bool upd = (inter * bu) > (bi * u);     // inter/u > bi/bu, strict => first argmax
        bi   = upd ? inter : bi;
        bu   = upd ? u     : bu;
        bidx = upd ? j     : bidx;
    }

    const bool matched = bi > 0.5f * bu;        // max_iou > IOU_THRESHOLD
    const float label  = matched ? 1.0f : 0.0f;

    // squared error vs matched GT (raw xywh space)
    float4 g = rawGT[bidx];
    float dx = pb.x - g.x, dy = pb.y - g.y, dz = pb.z - g.z, dw = pb.w - g.w;
    float e = matched ? (dx*dx + dy*dy + dz*dz + dw*dw) : 0.0f;

    // BCE-with-logits: softplus(s) - s*label   (numerically stable)
    float sp = fmaxf(s, 0.0f) + log1pf(__expf(-fabsf(s)));
    float b  = sp - s * label;
    float c  = label;

    // ---- wave32 reduction ----
#pragma unroll
    for (int m = 16; m >= 1; m >>= 1) {
        e += __shfl_xor(e, m, 32);
        c += __shfl_xor(c, m, 32);
        b += __shfl_xor(b, m, 32);
    }
    const int wave = tid >> 5, lane = tid & 31;
    if (lane == 0) { redE[wave] = e; redC[wave] = c; redB[wave] = b; }
    __syncthreads();
    if (tid == 0) {
        float te = 0.0f, tc = 0.0f, tb = 0.0f;
#pragma unroll
        for (int w = 0; w < 8; ++w) { te += redE[w]; tc += redC[w]; tb += redB[w]; }
        atomicAdd(&acc[img * 3 + 0], te);
        atomicAdd(&acc[img * 3 + 1], tc);
        atomicAdd(&acc[img * 3 + 2], tb);
    }
}

// ---------------------------------------------------------------------------
// Kernel 2: per-image loss + mean over batch. One wave32.
//   Final 16-value sum via V_WMMA_F32_16X16X4_F32 (A = losses, B = ones).
//   Summing one full D column (rows 0..15 via lane0/lane16) equals the sum of
//   every element of A, independent of A's exact VGPR striping.
// ---------------------------------------------------------------------------
__global__ void yolo_final(const float* __restrict__ acc, float* __restrict__ out) {
    const int L = threadIdx.x;
    float a0 = 0.0f;
    if (L < BATCH) {
        float err = acc[L * 3 + 0];
        float cnt = acc[L * 3 + 1];
        float bce = acc[L * 3 + 2];
        float loc = (cnt > 0.0f) ? (err / (4.0f * cnt)) : 0.0f;
        float ob  = bce * (1.0f / (float)N_PRED);
        float pen = 0.1f * ((float)N_PRED - cnt);
        a0 = loc + ob + pen;
    }
    v2f A = {a0, 0.0f};
    v2f B = {1.0f, 1.0f};
    v8f C = {0.f, 0.f, 0.f, 0.f, 0.f, 0.f, 0.f, 0.f};
    v8f D = __builtin_amdgcn_wmma_f32_16x16x4_f32(
        /*neg_a=*/false, A, /*neg_b=*/false, B,
        /*c_mod=*/(short)0, C, /*reuse_a=*/false, /*reuse_b=*/false);
    float part = D[0] + D[1] + D[2] + D[3] + D[4] + D[5] + D[6] + D[7]; // rows 0..7 (lane<16)
    float tot  = part + __shfl(part, 16, 32);                           // + rows 8..15
    if (L == 0) out[0] = tot * (1.0f / (float)BATCH);
}

// ---------------------------------------------------------------------------
extern "C" void kernel_launch(void* const* d_in, const int* in_sizes, int n_in,
                              void* d_out, int out_size, void* d_ws, size_t ws_size,
                              hipStream_t stream) {
    const float* y_hat = (const float*)d_in[0];  // (16, 8192, 4) f32
    const float* y     = (const float*)d_in[1];  // (16, 256, 4)  f32
    const float* obj   = (const float*)d_in[2];  // (16, 8192)    f32
    float* out = (float*)d_out;                  // scalar f32
    float* acc = (float*)d_ws;                   // 48 floats

    yolo_zero_ws<<<1, 64, 0, stream>>>(acc, BATCH * 3);
    yolo_main<<<BATCH * BLOCKS_PER_IMG, TPB, 0, stream>>>(y_hat, y, obj, acc);
    yolo_final<<<1, 32, 0, stream>>>(acc, out);
}